// SAGEEncoder_ov_69475390980563
// MI455X (gfx1250) — compile-verified
//
#include <hip/hip_runtime.h>

#define N_NODES 50000
#define N_EDGES 800000

typedef __attribute__((ext_vector_type(2))) float v2f;
typedef __attribute__((ext_vector_type(8))) float v8f;

// ---------------------------------------------------------------- utilities

__global__ void fill_zero_kernel(float* __restrict__ p, long n) {
  long i = (long)blockIdx.x * blockDim.x + threadIdx.x;
  if (i < n) p[i] = 0.0f;
}

__global__ void deg_kernel(const long long* __restrict__ dst,
                           float* __restrict__ deg, int nE) {
  int e = blockIdx.x * blockDim.x + threadIdx.x;
  if (e < nE) atomicAdd(&deg[(int)dst[e]], 1.0f);
}

__global__ void invdeg_kernel(float* __restrict__ deg, int n) {
  int i = blockIdx.x * blockDim.x + threadIdx.x;
  if (i < n) deg[i] = 1.0f / fmaxf(deg[i], 1.0f);
}

// Mean aggregation in one pass: agg[dst] += feat[src] * (1/deg[dst]).
// (1/deg[dst] is an edge-constant, so scaling before the atomic sum is exact
// up to fp reassociation — removes the per-WMMA multiply from the GEMM.)
__global__ void scatter_mean_kernel(const float* __restrict__ feat,
                                    const long long* __restrict__ src,
                                    const long long* __restrict__ dst,
                                    const float* __restrict__ inv,
                                    float* __restrict__ agg,
                                    int nE, int ld, int chunks) {
  long t = (long)blockIdx.x * blockDim.x + threadIdx.x;
  long e = t / chunks;
  int c = (int)(t - e * chunks);
  if (e >= nE) return;
  int s = (int)src[e];
  int d = (int)dst[e];
  const float invd = inv[d];
  const float4 v = *(const float4*)(feat + (long)s * ld + (long)c * 4);
  float* p = agg + (long)d * ld + (long)c * 4;
  atomicAdd(p + 0, v.x * invd);
  atomicAdd(p + 1, v.y * invd);
  atomicAdd(p + 2, v.z * invd);
  atomicAdd(p + 3, v.w * invd);
}

// --------------------------------------------------------------- SAGE GEMM
// out[M, Nld] = act( Aagg @ Wl + Ax @ Wr + b ) via V_WMMA_F32_16X16X4_F32.
// Aagg is the pre-normalized neighbor mean. Two 16-col output tiles per pass
// (two independent accumulator chains, A fragments loaded once per pass).
// A buffers have padded stride Kp with zero padding; W is [Kact,Nact] row-major.
#define TILE_STRIDE (128 * 16)  // LDS floats per 16-col weight tile (Kp <= 128)

__global__ __launch_bounds__(128)
void sage_gemm_kernel(const float* __restrict__ Aagg,
                      const float* __restrict__ Ax,
                      const float* __restrict__ Wl,
                      const float* __restrict__ Wr,
                      const float* __restrict__ bias,
                      float* __restrict__ out,
                      int M, int Kact, int Kp, int Nact, int Nld, int doRelu) {
  __shared__ float ldsWl[2 * TILE_STRIDE];   // 16 KB: Wl tile0 | Wl tile1
  __shared__ float ldsWr[2 * TILE_STRIDE];   // 16 KB: Wr tile0 | Wr tile1

  const int tid  = threadIdx.x;
  const int wave = tid >> 5;
  const int lane = tid & 31;
  const int half = lane >> 4;           // 0: lanes 0-15, 1: lanes 16-31
  const int nl   = lane & 15;

  const int rowBase = (blockIdx.x * 4 + wave) * 16;
  const int rowA    = rowBase + nl;     // A-matrix: M = lane & 15
  const int rowAc   = rowA < M ? rowA : M - 1;
  const long aBase  = (long)rowAc * Kp;

  const int nTiles  = (Nld + 15) >> 4;
  const int nPasses = (nTiles + 1) >> 1;
  const int wElems  = Kp * 32;          // 2 tiles worth of one weight matrix

  for (int pass = 0; pass < nPasses; ++pass) {
    const int colBase = pass << 5;      // 32 columns per pass

    // Cooperative stage of 4 weight tiles (Wl/Wr x 2 col-tiles), zero padded.
    for (int i = tid; i < wElems; i += 128) {
      int k = i >> 5, c = i & 31;
      int col = colBase + c;
      int idx = (c >> 4) * TILE_STRIDE + k * 16 + (c & 15);
      bool ok = (k < Kact) && (col < Nact);
      ldsWl[idx] = ok ? Wl[(long)k * Nact + col] : 0.0f;
      ldsWr[idx] = ok ? Wr[(long)k * Nact + col] : 0.0f;
    }
    __syncthreads();

    v8f acc0 = {};
    v8f acc1 = {};
    for (int k0 = 0; k0 < Kp; k0 += 4) {
      const int ka = k0 + (half << 1);  // lanes 0-15: k0,k0+1; 16-31: k0+2,k0+3
      const int bo = ka * 16 + nl;
      v2f a0, a1, b;

      a0.x = Aagg[aBase + ka];          // neighbor-mean fragment
      a0.y = Aagg[aBase + ka + 1];
      a1.x = Ax[aBase + ka];            // root (self) fragment
      a1.y = Ax[aBase + ka + 1];

      b.x = ldsWl[bo];
      b.y = ldsWl[bo + 16];
      acc0 = __builtin_amdgcn_wmma_f32_16x16x4_f32(false, a0, false, b,
                                                   (short)0, acc0, false, false);
      b.x = ldsWr[bo];
      b.y = ldsWr[bo + 16];
      acc0 = __builtin_amdgcn_wmma_f32_16x16x4_f32(false, a1, false, b,
                                                   (short)0, acc0, false, false);
      b.x = ldsWl[TILE_STRIDE + bo];
      b.y = ldsWl[TILE_STRIDE + bo + 16];
      acc1 = __builtin_amdgcn_wmma_f32_16x16x4_f32(false, a0, false, b,
                                                   (short)0, acc1, false, false);
      b.x = ldsWr[TILE_STRIDE + bo];
      b.y = ldsWr[TILE_STRIDE + bo + 16];
      acc1 = __builtin_amdgcn_wmma_f32_16x16x4_f32(false, a1, false, b,
                                                   (short)0, acc1, false, false);
    }
    __syncthreads();

    // C/D layout: VGPR v, lanes 0-15 -> M=v, lanes 16-31 -> M=v+8; N = lane&15
#pragma unroll
    for (int v = 0; v < 8; ++v) {
      int row = rowBase + v + (half << 3);
      if (row >= M) continue;
#pragma unroll
      for (int t = 0; t < 2; ++t) {
        int col = colBase + (t << 4) + nl;
        if (col < Nld) {
          float val = 0.0f;
          if (col < Nact) {
            val = (t ? acc1[v] : acc0[v]) + bias[col];
            if (doRelu) val = fmaxf(val, 0.0f);
          }
          out[(long)row * Nld + col] = val;
        }
      }
    }
  }
}

// ------------------------------------------------------------------ driver

extern "C" void kernel_launch(void* const* d_in, const int* in_sizes, int n_in,
                              void* d_out, int out_size, void* d_ws, size_t ws_size,
                              hipStream_t stream) {
  const float*     x   = (const float*)d_in[0];
  const long long* ei  = (const long long*)d_in[1];
  const long long* src = ei;             // edge_index[0, :]
  const long long* dst = ei + N_EDGES;   // edge_index[1, :]
  const float* Wl1 = (const float*)d_in[2];
  const float* Wr1 = (const float*)d_in[3];
  const float* b1  = (const float*)d_in[4];
  const float* Wl2 = (const float*)d_in[5];
  const float* Wr2 = (const float*)d_in[6];
  const float* b2  = (const float*)d_in[7];
  const float* Wlm = (const float*)d_in[8];
  const float* Wrm = (const float*)d_in[9];
  const float* bm  = (const float*)d_in[10];
  const float* Wlv = (const float*)d_in[11];
  const float* Wrv = (const float*)d_in[12];
  const float* bv  = (const float*)d_in[13];
  float* out = (float*)d_out;

  // Workspace layout (floats): deg/invdeg | agg(128) | h1(128) | h2(48)
  float* ws  = (float*)d_ws;
  float* deg = ws;                                 // N_NODES
  float* agg = deg + N_NODES;                      // N_NODES*128
  float* h1  = agg + (long)N_NODES * 128;          // N_NODES*128
  float* h2  = h1  + (long)N_NODES * 128;          // N_NODES*48

  const int T = 256;
  const long aggN128 = (long)N_NODES * 128;
  const long aggN48  = (long)N_NODES * 48;
  const int  gemmGrid = (N_NODES + 63) / 64;       // 4 waves/block, 16 rows/wave

  // --- degree -> 1/max(deg,1), shared by all layers
  fill_zero_kernel<<<(N_NODES + T - 1) / T, T, 0, stream>>>(deg, N_NODES);
  deg_kernel<<<(N_EDGES + T - 1) / T, T, 0, stream>>>(dst, deg, N_EDGES);
  invdeg_kernel<<<(N_NODES + T - 1) / T, T, 0, stream>>>(deg, N_NODES);

  // --- layer 1: h1 = relu(mean_agg(x) @ Wl1 + x @ Wr1 + b1)   [117 -> pad 128]
  fill_zero_kernel<<<(int)((aggN128 + T - 1) / T), T, 0, stream>>>(agg, aggN128);
  scatter_mean_kernel<<<(int)(((long)N_EDGES * 32 + T - 1) / T), T, 0, stream>>>(
      x, src, dst, deg, agg, N_EDGES, 128, 32);
  sage_gemm_kernel<<<gemmGrid, 128, 0, stream>>>(
      agg, x, Wl1, Wr1, b1, h1, N_NODES, 128, 128, 117, 128, 1);

  // --- layer 2: h2 = relu(mean_agg(h1) @ Wl2 + h1 @ Wr2 + b2) [42 -> pad 48]
  fill_zero_kernel<<<(int)((aggN128 + T - 1) / T), T, 0, stream>>>(agg, aggN128);
  scatter_mean_kernel<<<(int)(((long)N_EDGES * 32 + T - 1) / T), T, 0, stream>>>(
      h1, src, dst, deg, agg, N_EDGES, 128, 32);
  sage_gemm_kernel<<<gemmGrid, 128, 0, stream>>>(
      agg, h1, Wl2, Wr2, b2, h2, N_NODES, 117, 128, 42, 48, 1);

  // --- heads: one aggregation of h2, reused by mu and logvar
  fill_zero_kernel<<<(int)((aggN48 + T - 1) / T), T, 0, stream>>>(agg, aggN48);
  scatter_mean_kernel<<<(int)(((long)N_EDGES * 12 + T - 1) / T), T, 0, stream>>>(
      h2, src, dst, deg, agg, N_EDGES, 48, 12);
  sage_gemm_kernel<<<gemmGrid, 128, 0, stream>>>(
      agg, h2, Wlm, Wrm, bm, out, N_NODES, 42, 48, 24, 24, 0);
  sage_gemm_kernel<<<gemmGrid, 128, 0, stream>>>(
      agg, h2, Wlv, Wrv, bv, out + (long)N_NODES * 24,
      N_NODES, 42, 48, 24, 24, 0);
}